// PatchGAT_146028888216
// MI455X (gfx1250) — compile-verified
//
#include <hip/hip_runtime.h>

typedef float v2f __attribute__((ext_vector_type(2)));
typedef float v8f __attribute__((ext_vector_type(8)));

static constexpr float kNegSlope = 0.2f;

// ---------------------------------------------------------------------------
// C[M x ncols] = A[M x K] * B[K x ncols], all row-major fp32, via
// V_WMMA_F32_16X16X4_F32.  One wave (32 threads) per 16-row x (16*CT)-col
// block: each A load feeds CT WMMAs (A reuse across column tiles).
// Requires M % 16 == 0, ncols % (16*CT) == 0, K % 4 == 0 (all true here).
// A 16x4 tile layout: lanes 0-15 -> M=lane, K={0,1}; lanes 16-31 -> K={2,3}.
// B 4x16 tile layout mirrors it (row striped across lanes within a VGPR).
// C/D: VGPR v, lanes 0-15 -> (M=v, N=lane); lanes 16-31 -> (M=v+8, N=lane-16).
// ---------------------------------------------------------------------------
template <int CT>
__global__ __launch_bounds__(32)
void gat_gemm_wmma_f32(const float* __restrict__ A, const float* __restrict__ B,
                       float* __restrict__ C, int K, int ncols)
{
    const int lane    = threadIdx.x;
    const int half    = lane >> 4;        // 0 or 1
    const int l15     = lane & 15;
    const int kk      = half * 2;         // K sub-offset of this lane's pair
    const int colBase = blockIdx.x * (16 * CT);
    const int rowBase = blockIdx.y * 16;

    const float* __restrict__ Arow = A + (size_t)(rowBase + l15) * K;

    v8f acc[CT];
#pragma unroll
    for (int t = 0; t < CT; ++t) acc[t] = v8f{};

    for (int k0 = 0; k0 < K; k0 += 4) {
        v2f a;
        a.x = Arow[k0 + kk];
        a.y = Arow[k0 + kk + 1];
        const float* __restrict__ Brow0 = B + (size_t)(k0 + kk)     * ncols + colBase + l15;
        const float* __restrict__ Brow1 = B + (size_t)(k0 + kk + 1) * ncols + colBase + l15;
#pragma unroll
        for (int t = 0; t < CT; ++t) {
            v2f b;
            b.x = Brow0[t * 16];
            b.y = Brow1[t * 16];
            acc[t] = __builtin_amdgcn_wmma_f32_16x16x4_f32(
                /*neg_a=*/false, a, /*neg_b=*/false, b,
                /*c_mod=*/(short)0, acc[t], /*reuse_a=*/false, /*reuse_b=*/false);
        }
    }

#pragma unroll
    for (int t = 0; t < CT; ++t) {
#pragma unroll
        for (int v = 0; v < 8; ++v) {
            const int row = rowBase + v + half * 8;
            C[(size_t)row * ncols + colBase + t * 16 + l15] = acc[t][v];
        }
    }
}

// ---------------------------------------------------------------------------
// Utility fills (scratch is poisoned by the harness; re-init every call).
// ---------------------------------------------------------------------------
__global__ void fill_f32(float* __restrict__ p, float v, long long n)
{
    long long i = (long long)blockIdx.x * blockDim.x + threadIdx.x;
    if (i < n) p[i] = v;
}

__global__ void fill_u32(unsigned* __restrict__ p, unsigned v, long long n)
{
    long long i = (long long)blockIdx.x * blockDim.x + threadIdx.x;
    if (i < n) p[i] = v;
}

// ---------------------------------------------------------------------------
// Per-node attention coefficients: asrc[n,h] = <h[n,h,:], att_src[h,:]>, etc.
// ---------------------------------------------------------------------------
__global__ void gat_att_coeff(const float* __restrict__ h,
                              const float* __restrict__ att_src,
                              const float* __restrict__ att_dst,
                              float* __restrict__ asrc, float* __restrict__ adst,
                              int N, int H, int C)
{
    int t = blockIdx.x * blockDim.x + threadIdx.x;   // over N*H
    if (t >= N * H) return;
    int n = t / H, hd = t - n * H;
    const float* __restrict__ hp = h + (size_t)n * H * C + (size_t)hd * C;
    const float* __restrict__ as = att_src + hd * C;
    const float* __restrict__ ad = att_dst + hd * C;
    float s = 0.f, d = 0.f;
    for (int c = 0; c < C; ++c) { float v = hp[c]; s += v * as[c]; d += v * ad[c]; }
    asrc[t] = s; adst[t] = d;
}

// Monotone float -> uint mapping so unsigned atomicMax orders like floats.
__device__ __forceinline__ unsigned f32_ord(float f)
{
    unsigned u = __float_as_uint(f);
    return (u & 0x80000000u) ? ~u : (u | 0x80000000u);
}
__device__ __forceinline__ float ord_f32(unsigned o)
{
    unsigned u = (o & 0x80000000u) ? (o & 0x7FFFFFFFu) : ~o;
    return __uint_as_float(u);
}

// ---------------------------------------------------------------------------
// Pass 1 over messages (E edges + N self loops): leaky-relu score + seg max.
// ---------------------------------------------------------------------------
__global__ void gat_edge_score(const int* __restrict__ ei, long long E, long long M,
                               const float* __restrict__ asrc,
                               const float* __restrict__ adst,
                               float* __restrict__ ebuf, unsigned* __restrict__ mmax,
                               int H)
{
    long long m = (long long)blockIdx.x * blockDim.x + threadIdx.x;
    if (m >= M) return;
    const int src = (m < E) ? ei[m]     : (int)(m - E);
    const int dst = (m < E) ? ei[E + m] : (int)(m - E);
    for (int h = 0; h < H; ++h) {
        float e = asrc[(size_t)src * H + h] + adst[(size_t)dst * H + h];
        e = (e > 0.f) ? e : kNegSlope * e;
        ebuf[m * H + h] = e;
        atomicMax(&mmax[(size_t)dst * H + h], f32_ord(e));
    }
}

// Pass 2: ex = exp(e - max[dst]);  denom[dst] += ex.
__global__ void gat_edge_exp(const int* __restrict__ ei, long long E, long long M,
                             const unsigned* __restrict__ mmax,
                             float* __restrict__ ebuf, float* __restrict__ denom,
                             int H)
{
    long long m = (long long)blockIdx.x * blockDim.x + threadIdx.x;
    if (m >= M) return;
    const int dst = (m < E) ? ei[E + m] : (int)(m - E);
    for (int h = 0; h < H; ++h) {
        float mx = ord_f32(mmax[(size_t)dst * H + h]);
        float ex = __expf(ebuf[m * H + h] - mx);
        ebuf[m * H + h] = ex;
        atomicAdd(&denom[(size_t)dst * H + h], ex);
    }
}

// Pass 3: normalize in place: alpha = ex / denom[dst].
__global__ void gat_edge_alpha(const int* __restrict__ ei, long long E, long long M,
                               const float* __restrict__ denom,
                               float* __restrict__ ebuf, int H)
{
    long long m = (long long)blockIdx.x * blockDim.x + threadIdx.x;
    if (m >= M) return;
    const int dst = (m < E) ? ei[E + m] : (int)(m - E);
    for (int h = 0; h < H; ++h)
        ebuf[m * H + h] = ebuf[m * H + h] / denom[(size_t)dst * H + h];
}

// Pass 4: agg[dst, c..c+3] += alpha * h[src, c..c+3].  One thread per
// (message, 4 channels): one b128 gather + four f32 global atomic adds.
// This gather/atomic stream is the roofline of the whole workload.
__global__ void gat_edge_aggregate4(const int* __restrict__ ei, long long E, long long M,
                                    const float4* __restrict__ hfeat4,
                                    const float* __restrict__ alpha,
                                    float* __restrict__ agg, int H, int C4)
{
    const int HC4 = H * C4;                 // quads per node
    long long t = (long long)blockIdx.x * blockDim.x + threadIdx.x;
    long long total = M * (long long)HC4;
    if (t >= total) return;
    long long m = t / HC4;
    int q  = (int)(t - m * HC4);            // quad index within node features
    int hd = q / C4;
    const int src = (m < E) ? ei[m]     : (int)(m - E);
    const int dst = (m < E) ? ei[E + m] : (int)(m - E);
    float  al = alpha[m * H + hd];
    float4 v  = hfeat4[(size_t)src * HC4 + q];
    float* __restrict__ d = agg + ((size_t)dst * HC4 + q) * 4;
    atomicAdd(d + 0, al * v.x);
    atomicAdd(d + 1, al * v.y);
    atomicAdd(d + 2, al * v.z);
    atomicAdd(d + 3, al * v.w);
}

// Finalize: out = agg + bias, optional ReLU.
__global__ void gat_bias_act(const float* __restrict__ agg,
                             const float* __restrict__ bias,
                             float* __restrict__ out, long long n, int F, int relu)
{
    long long t = (long long)blockIdx.x * blockDim.x + threadIdx.x;
    if (t >= n) return;
    float v = agg[t] + bias[(int)(t % F)];
    out[t] = relu ? fmaxf(v, 0.f) : v;
}

// ---------------------------------------------------------------------------
// Host-side orchestration
// ---------------------------------------------------------------------------
static inline int blocks_for(long long n, int bs) { return (int)((n + bs - 1) / bs); }

extern "C" void kernel_launch(void* const* d_in, const int* in_sizes, int n_in,
                              void* d_out, int out_size, void* d_ws, size_t ws_size,
                              hipStream_t stream) {
    (void)n_in; (void)out_size; (void)ws_size;
    const float* x    = (const float*)d_in[0];
    const int*   ei   = (const int*)  d_in[1];
    const float* W1   = (const float*)d_in[2];
    const float* as1  = (const float*)d_in[3];
    const float* ad1  = (const float*)d_in[4];
    const float* b1   = (const float*)d_in[5];
    const float* W2   = (const float*)d_in[6];
    const float* as2  = (const float*)d_in[7];
    const float* ad2  = (const float*)d_in[8];
    const float* b2   = (const float*)d_in[9];
    float*       out  = (float*)d_out;

    const int  Fin = 256, H1 = 2, C1 = 64, HC1 = H1 * C1;   // 128
    const int  H2  = 1,  C2 = 64, HC2 = H2 * C2;            // 64
    const long long N = in_sizes[0] / Fin;                  // 50000 (mult of 16)
    const long long E = in_sizes[1] / 2;
    const long long M = E + N;                              // messages incl. self loops

    // Carve workspace (floats); all offsets are multiples of 4 floats -> 16B ok.
    float* ws     = (float*)d_ws;
    float* h1     = ws;             ws += N * HC1;
    float* asrc1  = ws;             ws += N * H1;
    float* adst1  = ws;             ws += N * H1;
    float* e1     = ws;             ws += M * H1;
    unsigned* m1  = (unsigned*)ws;  ws += N * H1;
    float* den1   = ws;             ws += N * H1;
    float* agg1   = ws;             ws += N * HC1;
    float* h1r    = ws;             ws += N * HC1;
    float* h2     = ws;             ws += N * HC2;
    float* asrc2  = ws;             ws += N * H2;
    float* adst2  = ws;             ws += N * H2;
    float* e2     = ws;             ws += M * H2;
    unsigned* m2  = (unsigned*)ws;  ws += N * H2;
    float* den2   = ws;             ws += N * H2;
    float* agg2   = ws;             ws += N * HC2;

    const int BS = 256;

    // ----------------- Layer 1 -----------------
    {
        dim3 grid(HC1 / 64, (unsigned)(N / 16));            // CT=4 -> 64 cols/wave
        gat_gemm_wmma_f32<4><<<grid, 32, 0, stream>>>(x, W1, h1, Fin, HC1);
    }
    gat_att_coeff<<<blocks_for(N * H1, BS), BS, 0, stream>>>(h1, as1, ad1, asrc1, adst1,
                                                             (int)N, H1, C1);
    fill_u32<<<blocks_for(N * H1, BS), BS, 0, stream>>>(m1, 0u, N * H1);
    fill_f32<<<blocks_for(N * H1, BS), BS, 0, stream>>>(den1, 0.f, N * H1);
    fill_f32<<<blocks_for(N * HC1, BS), BS, 0, stream>>>(agg1, 0.f, N * HC1);

    gat_edge_score<<<blocks_for(M, BS), BS, 0, stream>>>(ei, E, M, asrc1, adst1, e1, m1, H1);
    gat_edge_exp  <<<blocks_for(M, BS), BS, 0, stream>>>(ei, E, M, m1, e1, den1, H1);
    gat_edge_alpha<<<blocks_for(M, BS), BS, 0, stream>>>(ei, E, M, den1, e1, H1);
    gat_edge_aggregate4<<<blocks_for(M * (HC1 / 4), BS), BS, 0, stream>>>(
        ei, E, M, (const float4*)h1, e1, agg1, H1, C1 / 4);
    gat_bias_act<<<blocks_for(N * HC1, BS), BS, 0, stream>>>(agg1, b1, h1r, N * HC1, HC1, 1);

    // ----------------- Layer 2 -----------------
    {
        dim3 grid(HC2 / 64, (unsigned)(N / 16));            // CT=4 -> 64 cols/wave
        gat_gemm_wmma_f32<4><<<grid, 32, 0, stream>>>(h1r, W2, h2, HC1, HC2);
    }
    gat_att_coeff<<<blocks_for(N * H2, BS), BS, 0, stream>>>(h2, as2, ad2, asrc2, adst2,
                                                             (int)N, H2, C2);
    fill_u32<<<blocks_for(N * H2, BS), BS, 0, stream>>>(m2, 0u, N * H2);
    fill_f32<<<blocks_for(N * H2, BS), BS, 0, stream>>>(den2, 0.f, N * H2);
    fill_f32<<<blocks_for(N * HC2, BS), BS, 0, stream>>>(agg2, 0.f, N * HC2);

    gat_edge_score<<<blocks_for(M, BS), BS, 0, stream>>>(ei, E, M, asrc2, adst2, e2, m2, H2);
    gat_edge_exp  <<<blocks_for(M, BS), BS, 0, stream>>>(ei, E, M, m2, e2, den2, H2);
    gat_edge_alpha<<<blocks_for(M, BS), BS, 0, stream>>>(ei, E, M, den2, e2, H2);
    gat_edge_aggregate4<<<blocks_for(M * (HC2 / 4), BS), BS, 0, stream>>>(
        ei, E, M, (const float4*)h2, e2, agg2, H2, C2 / 4);
    gat_bias_act<<<blocks_for(N * HC2, BS), BS, 0, stream>>>(agg2, b2, out, N * HC2, HC2, 0);
}